// SkipGramNegativeSampling_7481833029874
// MI455X (gfx1250) — compile-verified
//
#include <hip/hip_runtime.h>

// SkipGram negative-sampling loss, MI455X (gfx1250, wave32).
//
// Roofline: 1.89 GB gathered reads @ 23.3 TB/s => ~81us floor; 0.87 GFLOP useful.
// Memory-bound, so the kernel streams each embedding row exactly once with
// global_load_b64 in *native WMMA operand layout* and lets the matrix pipes
// (v_wmma_f32_16x16x4_f32) do all dot products for a 16-sample tile per wave,
// reading the diagonal of the 16x16 product. 16x compute waste is free here.

#define EMBED  300
#define BATCH  131072
#define NEG    10
#define NTGT   (NEG + 1)     // context + 10 negatives
#define KSTEPS (EMBED / 4)   // 75 WMMA k-steps of 4
#define WAVES_PER_BLOCK 8    // 256 threads, 8 tiles of 16 samples per block
#define SAMPLES_PER_BLOCK (WAVES_PER_BLOCK * 16)
#define NBLOCKS (BATCH / SAMPLES_PER_BLOCK)  // 1024

typedef __attribute__((ext_vector_type(2))) float v2f;
typedef __attribute__((ext_vector_type(8))) float v8f;

__device__ __forceinline__ float log_sigmoid(float x) {
    // log(sigmoid(x)) = min(x,0) - log1p(exp(-|x|))   (numerically stable)
    return fminf(x, 0.0f) - log1pf(expf(-fabsf(x)));
}

__global__ __launch_bounds__(256, 1) void sgns_wmma_kernel(
    const int*   __restrict__ center_words,
    const int*   __restrict__ context_words,
    const int*   __restrict__ neg_samples,
    const float* __restrict__ in_embed,
    const float* __restrict__ out_embed,
    float*       __restrict__ block_partials)
{
    const int lane = threadIdx.x & 31;
    const int wave = threadIdx.x >> 5;
    const int tile = blockIdx.x * WAVES_PER_BLOCK + wave; // 16 samples per tile
    const int s    = lane & 15;   // sample within tile handled by this lane
    const int h    = lane >> 4;   // half-wave selects dim pair (k..k+1 vs k+2..k+3)
    const int gs   = tile * 16 + s;

    // Per-lane row base pointers. Lane L serves sample L%16 for both the
    // A operand (target rows) and the B operand (center row): for
    // V_WMMA_F32_16X16X4_F32 both operands want, in lane L, the two
    // consecutive f32 at dims (4j + 2*(L>>4)) of row (L&15) -> one b64 load.
    const float* rowB = in_embed + (long)center_words[gs] * EMBED;
    const float* rowA[NTGT];
    rowA[0] = out_embed + (long)context_words[gs] * EMBED;
#pragma unroll
    for (int t = 0; t < NEG; ++t)
        rowA[t + 1] = out_embed + (long)neg_samples[gs * NEG + t] * EMBED;

    v8f acc[NTGT];
#pragma unroll
    for (int t = 0; t < NTGT; ++t) acc[t] = {};

    const int dbase = 2 * h;
    for (int j = 0; j < KSTEPS; ++j) {
        const int d = 4 * j + dbase;
        v2f b = *(const v2f*)(rowB + d);          // center: B[k][n] layout
#pragma unroll
        for (int t = 0; t < NTGT; ++t) {
            v2f a = *(const v2f*)(rowA[t] + d);   // target: A[m][k] layout
            // (neg_a, A, neg_b, B, c_mod, C, reuse_a, reuse_b)
            acc[t] = __builtin_amdgcn_wmma_f32_16x16x4_f32(
                false, a, false, b, (short)0, acc[t], false, false);
        }
    }

    // Extract diagonal: C/D layout VGPR r: lanes 0-15 hold M=r (N=lane),
    // lanes 16-31 hold M=r+8 (N=lane-16).  Diagonal elements live at
    // (reg r, lane r) for sample r and (reg r, lane r+24) for sample r+8.
    float lsum = 0.0f;
#pragma unroll
    for (int t = 0; t < NTGT; ++t) {
        float sc = 0.0f;
#pragma unroll
        for (int r = 0; r < 8; ++r) {
            float a  = acc[t][r];
            float d0 = __shfl(a, r, 32);
            float d1 = __shfl(a, r + 24, 32);
            if (lane == r)     sc = d0;   // sample r
            if (lane == r + 8) sc = d1;   // sample r+8
        }
        lsum += (t == 0) ? log_sigmoid(sc) : log_sigmoid(-sc);
    }
    if (h != 0) lsum = 0.0f;  // only lanes 0-15 hold valid per-sample losses

    // Deterministic fixed-tree wave reduction (lanes 16-31 contribute 0).
#pragma unroll
    for (int off = 16; off >= 1; off >>= 1)
        lsum += __shfl_xor(lsum, off, 32);

    __shared__ float wsum[WAVES_PER_BLOCK];
    if (lane == 0) wsum[wave] = lsum;
    __syncthreads();
    if (threadIdx.x == 0) {
        float tot = 0.0f;
#pragma unroll
        for (int w = 0; w < WAVES_PER_BLOCK; ++w) tot += wsum[w];
        block_partials[blockIdx.x] = tot;
    }
}

__global__ __launch_bounds__(256, 1) void sgns_reduce_kernel(
    const float* __restrict__ partials, int n, float* __restrict__ out)
{
    __shared__ float sh[256];
    float s = 0.0f;
    for (int i = threadIdx.x; i < n; i += 256) s += partials[i];
    sh[threadIdx.x] = s;
    __syncthreads();
    for (int off = 128; off >= 1; off >>= 1) {
        if ((int)threadIdx.x < off) sh[threadIdx.x] += sh[threadIdx.x + off];
        __syncthreads();
    }
    if (threadIdx.x == 0) out[0] = -sh[0] / (float)BATCH;
}

extern "C" void kernel_launch(void* const* d_in, const int* in_sizes, int n_in,
                              void* d_out, int out_size, void* d_ws, size_t ws_size,
                              hipStream_t stream) {
    const int*   center  = (const int*)d_in[0];
    const int*   context = (const int*)d_in[1];
    const int*   negs    = (const int*)d_in[2];
    const float* in_e    = (const float*)d_in[3];
    const float* out_e   = (const float*)d_in[4];
    float* partials = (float*)d_ws;   // NBLOCKS floats, fully rewritten each call
    float* out      = (float*)d_out;

    sgns_wmma_kernel<<<dim3(NBLOCKS), dim3(256), 0, stream>>>(
        center, context, negs, in_e, out_e, partials);
    sgns_reduce_kernel<<<dim3(1), dim3(256), 0, stream>>>(partials, NBLOCKS, out);
}